// SuperpointGenerator_83940840833726
// MI455X (gfx1250) — compile-verified
//
#include <hip/hip_runtime.h>
#include <stdint.h>

// ---------------------------------------------------------------------------
// Superpoint label generation for MI455X (gfx1250, wave32).
//
// Roofline: compulsory traffic ~96 MB coords + 32 MB labels -> ~5.7 us at
// 23.3 TB/s. No matrix FLOPs anywhere, so WMMA is not applicable; the CDNA5
// features that pay are async global->LDS staging (ASYNCcnt/s_wait_asynccnt),
// LDS atomics, and an L2-resident hash table (keys+counts = 64 MB << 192 MB
// L2) so the O(N) build-phase atomics are on-die RMWs instead of a per-batch
// O(N log N) sort.
// ---------------------------------------------------------------------------

typedef unsigned int u32;
typedef unsigned long long u64;
typedef int v4i __attribute__((ext_vector_type(4)));

#ifndef __has_builtin
#define __has_builtin(x) 0
#endif

#if __has_builtin(__builtin_amdgcn_global_load_async_to_lds_b128)
#define HAVE_ASYNC_B128 1
#else
#define HAVE_ASYNC_B128 0
#endif
#if __has_builtin(__builtin_amdgcn_global_load_async_to_lds_b32)
#define HAVE_ASYNC_B32 1
#else
#define HAVE_ASYNC_B32 0
#endif

#define AS_GLOBAL __attribute__((address_space(1)))
#define AS_LDS    __attribute__((address_space(3)))

__device__ __forceinline__ void wait_async_zero() {
#if __has_builtin(__builtin_amdgcn_s_wait_asynccnt)
  __builtin_amdgcn_s_wait_asynccnt(0);
#else
  asm volatile("s_wait_asynccnt 0" ::: "memory");
#endif
}

static constexpr int kB      = 32;
static constexpr int kN      = 262144;     // points per batch (2^18)
static constexpr u32 kH      = 1u << 18;   // hash slots per batch (load ~0.3)
static constexpr u32 kHMask  = kH - 1u;
static constexpr int kMaxSP  = 512;
static constexpr int kTieCap = 4096;
static constexpr int kEmpty  = 0x7FFFFFFF; // |id| <= ~7e5, so this is safe
static constexpr int kNBuck  = 1024;       // count-histogram buckets

// ---------------------------------------------------------------------------
// K0: clear workspace (harness poisons ws; must re-init deterministically)
// ---------------------------------------------------------------------------
__global__ __launch_bounds__(256) void k_init(int* keys, u32* counts,
                                              u32* labels, u32* hist,
                                              u32* tval, u32* curg, u32* curt) {
  size_t gid = (size_t)blockIdx.x * 256 + threadIdx.x;  // over kB*kH
  keys[gid]   = kEmpty;
  counts[gid] = 0u;
  labels[gid] = 0u;
  if (gid < (size_t)kB * kNBuck) hist[gid] = 0u;
  if (gid < (size_t)kB) { tval[gid] = 1u; curg[gid] = 0u; curt[gid] = 0u; }
}

// ---------------------------------------------------------------------------
// K1: voxel-hash build. 256 threads = 8 waves; 256-point tile (3072 B) staged
// to LDS with CDNA5 async global->LDS (b128: threads 0..191 move 16 B each),
// then hash insert via atomicCAS + count via atomicAdd (both L2-resident).
// ---------------------------------------------------------------------------
__global__ __launch_bounds__(256) void k_build(const float* __restrict__ coords,
                                               int* keys, u32* counts,
                                               u32* slotp) {
  __shared__ float sc[256 * 3];
  const u32 tid  = threadIdx.x;
  const u32 tile = blockIdx.x & 1023u;        // kN/256 = 1024 tiles per batch
  const u32 b    = blockIdx.x >> 10;
  const size_t baseF = ((size_t)b * kN + (size_t)tile * 256) * 3;
  const float* gsrc = coords + baseF;

#if HAVE_ASYNC_B128
  // 3072 bytes per tile = 192 x 16B transfers (tile base is 3072B-aligned
  // from a 256B-aligned allocation, so every 16B chunk is aligned).
  if (tid < 192u) {
    __builtin_amdgcn_global_load_async_to_lds_b128(
        (AS_GLOBAL v4i*)(const_cast<float*>(gsrc) + tid * 4u),
        (AS_LDS v4i*)(&sc[tid * 4u]),
        /*imm offset*/ 0, /*cpol*/ 0);
  }
  wait_async_zero();
  __syncthreads();
#elif HAVE_ASYNC_B32
  for (int k = 0; k < 3; ++k) {
    const u32 d = tid + (u32)k * 256u;
    __builtin_amdgcn_global_load_async_to_lds_b32(
        (AS_GLOBAL int*)(const_cast<float*>(gsrc) + d),
        (AS_LDS int*)(&sc[d]),
        /*imm offset*/ 0, /*cpol*/ 0);
  }
  wait_async_zero();
  __syncthreads();
#else
  for (int k = 0; k < 3; ++k) sc[tid + k * 256] = gsrc[tid + k * 256];
  __syncthreads();
#endif

  // stride-3 LDS reads: 3 coprime with 64 banks -> conflict-free per wave32
  const float x = sc[tid * 3 + 0];
  const float y = sc[tid * 3 + 1];
  const float z = sc[tid * 3 + 2];
  // coords / 0.2 truncated toward zero == * 5.0 truncated (int cast truncates)
  const int vx = (int)(x * 5.0f);
  const int vy = (int)(y * 5.0f);
  const int vz = (int)(z * 5.0f);
  const int id = vx * 10000 + vy * 100 + vz;

  u32 h = (u32)id * 2654435761u;
  h ^= h >> 15;
  h &= kHMask;

  int* bk = keys + (size_t)b * kH;
  u32 slot = h;
  for (u32 p = 0; p < kH; ++p) {
    slot = (h + p) & kHMask;
    int prev = atomicCAS(&bk[slot], kEmpty, id);
    if (prev == kEmpty || prev == id) break;
  }
  atomicAdd(&counts[(size_t)b * kH + slot], 1u);
  slotp[(size_t)b * kN + (size_t)tile * 256 + tid] = slot;
}

// ---------------------------------------------------------------------------
// K2: per-batch histogram of per-voxel counts (clamped to kNBuck-1).
// LDS-local histogram (ds_add) then merged to global.
// ---------------------------------------------------------------------------
__global__ __launch_bounds__(256) void k_hist(const int* keys, const u32* counts,
                                              u32* hist) {
  __shared__ u32 lh[kNBuck];
  for (u32 i = threadIdx.x; i < (u32)kNBuck; i += 256) lh[i] = 0u;
  __syncthreads();

  const u32 b    = blockIdx.x >> 10;          // kH/256 = 1024 blocks per batch
  const u32 slot = ((blockIdx.x & 1023u) << 8) + threadIdx.x;
  const size_t idx = (size_t)b * kH + slot;
  if (keys[idx] != kEmpty) {
    u32 c = counts[idx];
    u32 cc = c < (u32)(kNBuck - 1) ? c : (u32)(kNBuck - 1);
    atomicAdd(&lh[cc], 1u);
  }
  __syncthreads();
  for (u32 i = threadIdx.x; i < (u32)kNBuck; i += 256)
    if (lh[i]) atomicAdd(&hist[(size_t)b * kNBuck + i], lh[i]);
}

// ---------------------------------------------------------------------------
// K3: find threshold T per batch = largest c with  #(count >= c) >= 512.
// Suffix inclusive scan over the 1024-bucket histogram (one WG per batch).
// ---------------------------------------------------------------------------
__global__ __launch_bounds__(1024) void k_thresh(const u32* hist, u32* tval) {
  __shared__ u32 s[kNBuck];
  const u32 c = threadIdx.x;
  const u32 b = blockIdx.x;
  s[c] = hist[(size_t)b * kNBuck + c];
  __syncthreads();
  for (u32 off = 1; off < (u32)kNBuck; off <<= 1) {
    u32 v = (c + off < (u32)kNBuck) ? s[c + off] : 0u;
    __syncthreads();
    s[c] += v;
    __syncthreads();
  }
  const u32 next = (c + 1 < (u32)kNBuck) ? s[c + 1] : 0u;
  if (c >= 1 && s[c] >= (u32)kMaxSP && next < (u32)kMaxSP) tval[b] = c;
  // if n_unique < 512: no thread fires, T stays at the preset 1.
}

// ---------------------------------------------------------------------------
// K4: collect winners (count > T) and boundary ties (count == T).
// By construction #(count>T) <= 511; ties capped at kTieCap.
// ---------------------------------------------------------------------------
__global__ __launch_bounds__(256) void k_collect(const int* keys, const u32* counts,
                                                 const u32* tval, u32* curg,
                                                 u32* curt, u64* winu, u32* wins,
                                                 u32* tieu, u32* ties) {
  const u32 b    = blockIdx.x >> 10;
  const u32 slot = ((blockIdx.x & 1023u) << 8) + threadIdx.x;
  const size_t idx = (size_t)b * kH + slot;
  const int id = keys[idx];
  if (id == kEmpty) return;
  const u32 cnt = counts[idx];
  const u32 cc  = cnt < (u32)(kNBuck - 1) ? cnt : (u32)(kNBuck - 1);
  const u32 T   = tval[b];
  const u32 uid = (u32)id ^ 0x80000000u;  // order-preserving int->uint
  if (cc > T) {
    u32 p = atomicAdd(&curg[b], 1u);
    if (p < (u32)kMaxSP) {
      // composite sorts (count desc, id asc) when taken descending
      winu[(size_t)b * kMaxSP + p] = ((u64)cnt << 32) | (u64)(0xFFFFFFFFu - uid);
      wins[(size_t)b * kMaxSP + p] = slot;
    }
  } else if (cc == T) {
    u32 p = atomicAdd(&curt[b], 1u);
    if (p < (u32)kTieCap) {
      tieu[(size_t)b * kTieCap + p] = uid;
      ties[(size_t)b * kTieCap + p] = slot;
    }
  }
}

// ---------------------------------------------------------------------------
// Bitonic ascending sort of key[0..n) with payload, in LDS, n power of two.
// ---------------------------------------------------------------------------
template <typename K>
__device__ void bitonic_asc(K* key, u32* val, u32 n, u32 tid, u32 nthr) {
  for (u32 k = 2; k <= n; k <<= 1) {
    for (u32 j = k >> 1; j > 0; j >>= 1) {
      __syncthreads();
      for (u32 i = tid; i < n; i += nthr) {
        const u32 ix = i ^ j;
        if (ix > i) {
          K a = key[i], c = key[ix];
          const bool up = ((i & k) == 0);
          if ((a > c) == up) {
            key[i] = c; key[ix] = a;
            u32 t = val[i]; val[i] = val[ix]; val[ix] = t;
          }
        }
      }
    }
  }
  __syncthreads();
}

// ---------------------------------------------------------------------------
// K5: per batch (one 1024-thread WG): sort ties by id, take the needed
// smallest, then sort the 512 winners by (count desc, id asc) and emit labels.
// ---------------------------------------------------------------------------
__global__ __launch_bounds__(1024) void k_final(const u32* counts, u32* labels,
                                                const u32* curg, const u32* curt,
                                                const u64* winu, const u32* wins,
                                                const u32* tieu, const u32* ties) {
  __shared__ u32 tk[kTieCap];
  __shared__ u32 tv[kTieCap];
  __shared__ u64 wk[kMaxSP];
  __shared__ u32 wv[kMaxSP];
  const u32 b   = blockIdx.x;
  const u32 tid = threadIdx.x;
  const u32 nG = curg[b] < (u32)kMaxSP ? curg[b] : (u32)kMaxSP;
  const u32 nT = curt[b] < (u32)kTieCap ? curt[b] : (u32)kTieCap;

  for (u32 i = tid; i < (u32)kTieCap; i += 1024) {
    tk[i] = (i < nT) ? tieu[(size_t)b * kTieCap + i] : 0xFFFFFFFFu;
    tv[i] = (i < nT) ? ties[(size_t)b * kTieCap + i] : 0xFFFFFFFFu;
  }
  __syncthreads();
  bitonic_asc<u32>(tk, tv, (u32)kTieCap, tid, 1024);  // ids ascending

  u32 need = (nG < (u32)kMaxSP) ? ((u32)kMaxSP - nG) : 0u;
  if (need > nT) need = nT;

  for (u32 i = tid; i < (u32)kMaxSP; i += 1024) {
    if (i < nG) {
      wk[i] = ~winu[(size_t)b * kMaxSP + i];  // sort ascending on ~u == desc on u
      wv[i] = wins[(size_t)b * kMaxSP + i];
    } else if (i < nG + need) {
      const u32 j = i - nG;
      const u32 slot = tv[j];
      const u32 cnt = counts[(size_t)b * kH + slot];
      const u64 u = ((u64)cnt << 32) | (u64)(0xFFFFFFFFu - tk[j]);
      wk[i] = ~u;
      wv[i] = slot;
    } else {
      wk[i] = ~0ull;           // pads sort last
      wv[i] = 0xFFFFFFFFu;     // sentinel (real slots < 2^18)
    }
  }
  __syncthreads();
  bitonic_asc<u64>(wk, wv, (u32)kMaxSP, tid, 1024);

  for (u32 i = tid; i < (u32)kMaxSP; i += 1024)
    if (wv[i] != 0xFFFFFFFFu)
      labels[(size_t)b * kH + wv[i]] = i;
}

// ---------------------------------------------------------------------------
// K6: gather per-point labels (int32 output).
// ---------------------------------------------------------------------------
__global__ __launch_bounds__(256) void k_out(const u32* __restrict__ slotp,
                                             const u32* __restrict__ labels,
                                             int* __restrict__ out) {
  const size_t gid = (size_t)blockIdx.x * 256 + threadIdx.x;  // over kB*kN
  const u32 b = (u32)(gid >> 18);  // kN == 2^18
  out[gid] = (int)labels[(size_t)b * kH + slotp[gid]];
}

// ---------------------------------------------------------------------------
extern "C" void kernel_launch(void* const* d_in, const int* in_sizes, int n_in,
                              void* d_out, int out_size, void* d_ws,
                              size_t ws_size, hipStream_t stream) {
  (void)in_sizes; (void)n_in; (void)out_size;
  const float* coords = (const float*)d_in[0];
  int* out = (int*)d_out;  // reference output dtype is int32

  char* ws = (char*)d_ws;
  size_t off = 0;
  auto carve = [&](size_t bytes) -> void* {
    void* p = ws + off;
    off = (off + bytes + 255) & ~(size_t)255;
    return p;
  };
  int* keys   = (int*)carve((size_t)kB * kH * 4);
  u32* counts = (u32*)carve((size_t)kB * kH * 4);
  u32* labels = (u32*)carve((size_t)kB * kH * 4);
  u32* slotp  = (u32*)carve((size_t)kB * kN * 4);
  u32* hist   = (u32*)carve((size_t)kB * kNBuck * 4);
  u32* tval   = (u32*)carve((size_t)kB * 4);
  u32* curg   = (u32*)carve((size_t)kB * 4);
  u32* curt   = (u32*)carve((size_t)kB * 4);
  u64* winu   = (u64*)carve((size_t)kB * kMaxSP * 8);
  u32* wins   = (u32*)carve((size_t)kB * kMaxSP * 4);
  u32* tieu   = (u32*)carve((size_t)kB * kTieCap * 4);
  u32* ties   = (u32*)carve((size_t)kB * kTieCap * 4);
  if (off > ws_size) return;  // workspace too small; bail deterministically

  const u32 gridTable  = (u32)((size_t)kB * kH / 256);   // 32768
  const u32 gridPoints = (u32)((size_t)kB * kN / 256);   // 32768

  k_init<<<gridTable, 256, 0, stream>>>(keys, counts, labels, hist, tval, curg,
                                        curt);
  k_build<<<gridPoints, 256, 0, stream>>>(coords, keys, counts, slotp);
  k_hist<<<gridTable, 256, 0, stream>>>(keys, counts, hist);
  k_thresh<<<kB, kNBuck, 0, stream>>>(hist, tval);
  k_collect<<<gridTable, 256, 0, stream>>>(keys, counts, tval, curg, curt, winu,
                                           wins, tieu, ties);
  k_final<<<kB, 1024, 0, stream>>>(counts, labels, curg, curt, winu, wins, tieu,
                                   ties);
  k_out<<<gridPoints, 256, 0, stream>>>(slotp, labels, out);
}